// DCT_base_Rec_Module_58677843198488
// MI455X (gfx1250) — compile-verified
//
#include <hip/hip_runtime.h>
#include <math.h>
#include <float.h>

// ---------------------------------------------------------------------------
// DCT_base_Rec_Module for MI455X (gfx1250, wave32).
// Heavy phase: per-patch 2D DCT (D*P*D^T) via V_WMMA_F32_16X16X4_F32 (fp32,
// matches reference precision) + grade reduction. Selection phase picks
// bottom-2 / top-2 grades per batch; output phase recomputes level_y for only
// the 4 selected patches per batch.
// ---------------------------------------------------------------------------

typedef __attribute__((ext_vector_type(2))) float v2f;
typedef __attribute__((ext_vector_type(8))) float v8f;

#define WSZ     32
#define LPITCH  36          // LDS row pitch in floats: conflict-free, 16B aligned

// --- WMMA helpers -----------------------------------------------------------
// Layouts assumed (per CDNA5 ISA 7.12.2):
//  A (16x4 f32):  lane<16 -> M=lane, v0=K0,v1=K1 ; lane>=16 -> v0=K2,v1=K3
//  B (4x16 f32):  lane<16 -> N=lane, v0=K0,v1=K1 ; lane>=16 -> v0=K2,v1=K3
//  C/D (16x16):   vgpr r -> M=r (lanes 0-15) / M=r+8 (lanes 16-31), N=lane%16

// One 16x16 output block of a 32x32 x 32x32 GEMM (K=32 -> 8 chained WMMAs).
// aT: read A as transpose of the stored matrix; bT likewise for B.
__device__ __forceinline__ v8f gemm16(const float* sA, int aRow0, bool aT,
                                      const float* sB, int bCol0, bool bT,
                                      int lane)
{
    const int half = lane >> 4;
    const int r    = lane & 15;     // row index for A, column index for B
    v8f c = {0.f,0.f,0.f,0.f,0.f,0.f,0.f,0.f};
    #pragma unroll
    for (int kq = 0; kq < 8; ++kq) {
        const int k0 = kq * 4 + half * 2;
        v2f a, b;
        if (!aT) {
            a.x = sA[(aRow0 + r) * LPITCH + k0];
            a.y = sA[(aRow0 + r) * LPITCH + k0 + 1];
        } else {
            a.x = sA[k0       * LPITCH + aRow0 + r];
            a.y = sA[(k0 + 1) * LPITCH + aRow0 + r];
        }
        if (!bT) {
            b.x = sB[k0       * LPITCH + bCol0 + r];
            b.y = sB[(k0 + 1) * LPITCH + bCol0 + r];
        } else {
            b.x = sB[(bCol0 + r) * LPITCH + k0];
            b.y = sB[(bCol0 + r) * LPITCH + k0 + 1];
        }
        c = __builtin_amdgcn_wmma_f32_16x16x4_f32(false, a, false, b,
                                                  (short)0, c, false, false);
    }
    return c;
}

__device__ __forceinline__ void gemm32(const float* sA, bool aT,
                                       const float* sB, bool bT,
                                       v8f acc[2][2], int lane)
{
    #pragma unroll
    for (int i = 0; i < 2; ++i)
        #pragma unroll
        for (int j = 0; j < 2; ++j)
            acc[i][j] = gemm16(sA, i * 16, aT, sB, j * 16, bT, lane);
}

// Scatter C/D-layout accumulators into a row-major LDS tile.
__device__ __forceinline__ void store_cd(float* sOut, v8f acc[2][2], int lane)
{
    const int half = lane >> 4;
    const int n    = lane & 15;
    #pragma unroll
    for (int i = 0; i < 2; ++i)
        #pragma unroll
        for (int j = 0; j < 2; ++j)
            #pragma unroll
            for (int rr = 0; rr < 8; ++rr)
                sOut[(i * 16 + half * 8 + rr) * LPITCH + j * 16 + n] = acc[i][j][rr];
}

// --- Kernel 0: build D (DCT matrix) and W (collapsed grade weights) ---------
__global__ __launch_bounds__(1024) void init_consts_kernel(float* __restrict__ Dg,
                                                           float* __restrict__ Wg)
{
    const int t = threadIdx.x;          // 1024 threads -> (i,j)
    const int i = t >> 5, j = t & 31;
    const double scale = (i == 0) ? sqrt(1.0 / 32.0) : sqrt(2.0 / 32.0);
    Dg[t] = (float)(scale * cos(((double)j + 0.5) * M_PI * (double)i / 32.0));

    // W[i][j] = sum_g in_filter(g, i+j) * 2^g / ft_num[g]
    const double s = (double)(i + j);
    double w = 0.0;
    for (int g = 0; g < 6; ++g) {
        const double start = 64.0 / 6.0 * (double)g;        // WS*2.0/GRADE_N*g
        const double end   = 64.0 / 6.0 * (double)(g + 1);
        double cnt = 0.0;
        for (int ss = 0; ss <= 62; ++ss) {                  // #cells with i+j==ss
            const double sd = (double)ss;
            if (!((sd > end) || (sd < start)))
                cnt += (double)((ss <= 31) ? ss + 1 : 63 - ss);
        }
        if (!((s > end) || (s < start)))
            w += exp2((double)g) / cnt;
    }
    Wg[t] = (float)w;
}

// --- Kernel 1: per-(b,l) DCT of all channels + grade -----------------------
__global__ __launch_bounds__(32) void dct_grade_kernel(
    const float* __restrict__ x, const float* __restrict__ Dg,
    const float* __restrict__ Wg, float* __restrict__ grade,
    int C, int H, int Wd, int nW, int L)
{
    __shared__ float sD[32 * LPITCH];
    __shared__ float sP[32 * LPITCH];
    __shared__ float sT[32 * LPITCH];

    const int lane = threadIdx.x;
    const int bl   = blockIdx.x;                 // b*L + l
    const int l    = bl % L;
    const int b    = bl / L;
    const int pr   = (l / nW) * 16;
    const int pc   = (l % nW) * 16;

    for (int t = lane; t < 1024; t += 32)
        sD[(t >> 5) * LPITCH + (t & 31)] = Dg[t];

    const int half = lane >> 4;
    const int n    = lane & 15;
    float gs = 0.f;

    for (int c = 0; c < C; ++c) {
        const float* xp = x + (((size_t)b * C + c) * H + pr) * (size_t)Wd + pc;
        __syncthreads();                         // WAR on sP/sT from prior iter
        for (int t = lane; t < 1024; t += 32)
            sP[(t >> 5) * LPITCH + (t & 31)] = xp[(size_t)(t >> 5) * Wd + (t & 31)];
        __syncthreads();

        v8f acc[2][2];
        gemm32(sD, false, sP, false, acc, lane); // T = D * P
        __syncthreads();
        store_cd(sT, acc, lane);
        __syncthreads();

        v8f xd[2][2];
        gemm32(sT, false, sD, true, xd, lane);   // X = T * D^T   (x_dct)

        #pragma unroll
        for (int i = 0; i < 2; ++i)
            #pragma unroll
            for (int j = 0; j < 2; ++j)
                #pragma unroll
                for (int rr = 0; rr < 8; ++rr) {
                    const int m  = i * 16 + half * 8 + rr;
                    const int nn = j * 16 + n;
                    const float v = xd[i][j][rr];
                    gs += logf(fabsf(v) + 1.0f) * Wg[m * 32 + nn];
                }
    }

    // wave32 reduction
    for (int off = 16; off > 0; off >>= 1)
        gs += __shfl_down(gs, off, 32);
    if (lane == 0) grade[bl] = gs;
}

// --- Kernel 2: bottom-2 / top-2 selection (stable-argsort semantics) -------
__device__ __forceinline__ void ins_min(float v, int i, float& m1, int& i1,
                                        float& m2, int& i2)
{
    if (i < 0) return;
    if (v < m1 || (v == m1 && i < i1)) { m2 = m1; i2 = i1; m1 = v; i1 = i; }
    else if (v < m2 || (v == m2 && i < i2)) { m2 = v; i2 = i; }
}
__device__ __forceinline__ void ins_max(float v, int i, float& m1, int& i1,
                                        float& m2, int& i2)
{
    if (i < 0) return;
    if (v > m1 || (v == m1 && i > i1)) { m2 = m1; i2 = i1; m1 = v; i1 = i; }
    else if (v > m2 || (v == m2 && i > i2)) { m2 = v; i2 = i; }
}

__global__ __launch_bounds__(256) void select_kernel(const float* __restrict__ grade,
                                                     int* __restrict__ sel, int L)
{
    const int b   = blockIdx.x;
    const int tid = threadIdx.x;
    const float* g = grade + (size_t)b * L;

    float mn1 = FLT_MAX, mn2 = FLT_MAX, mx1 = -FLT_MAX, mx2 = -FLT_MAX;
    int   mi1 = -1, mi2 = -1, xi1 = -1, xi2 = -1;
    for (int l = tid; l < L; l += 256) {
        const float v = g[l];
        ins_min(v, l, mn1, mi1, mn2, mi2);
        ins_max(v, l, mx1, xi1, mx2, xi2);
    }

    __shared__ float smn1[256], smn2[256], smx1[256], smx2[256];
    __shared__ int   smi1[256], smi2[256], sxi1[256], sxi2[256];
    smn1[tid] = mn1; smn2[tid] = mn2; smx1[tid] = mx1; smx2[tid] = mx2;
    smi1[tid] = mi1; smi2[tid] = mi2; sxi1[tid] = xi1; sxi2[tid] = xi2;
    __syncthreads();

    for (int off = 128; off > 0; off >>= 1) {
        if (tid < off) {
            const int p = tid + off;
            ins_min(smn1[p], smi1[p], mn1, mi1, mn2, mi2);
            ins_min(smn2[p], smi2[p], mn1, mi1, mn2, mi2);
            ins_max(smx1[p], sxi1[p], mx1, xi1, mx2, xi2);
            ins_max(smx2[p], sxi2[p], mx1, xi1, mx2, xi2);
        }
        __syncthreads();
        if (tid < off) {
            smn1[tid] = mn1; smn2[tid] = mn2; smx1[tid] = mx1; smx2[tid] = mx2;
            smi1[tid] = mi1; smi2[tid] = mi2; sxi1[tid] = xi1; sxi2[tid] = xi2;
        }
        __syncthreads();
    }

    if (tid == 0) {
        sel[b * 4 + 0] = mi1;   // minmin   (output slot 0)
        sel[b * 4 + 1] = xi1;   // maxmax   (output slot 1)
        sel[b * 4 + 2] = mi2;   // minmin1  (output slot 2)
        sel[b * 4 + 3] = xi2;   // maxmax1  (output slot 3)
    }
}

// --- Kernel 3: level_y for the 4 selected patches per batch ----------------
__global__ __launch_bounds__(32) void level_out_kernel(
    const float* __restrict__ x, const float* __restrict__ Dg,
    const int* __restrict__ sel, float* __restrict__ out,
    int B, int C, int H, int Wd, int nW)
{
    __shared__ float sD[32 * LPITCH];
    __shared__ float sP[32 * LPITCH];
    __shared__ float sT[32 * LPITCH];

    const int lane = threadIdx.x;
    int bid = blockIdx.x;                       // ((s*B + b)*C + c)
    const int c = bid % C;  bid /= C;
    const int b = bid % B;
    const int s = bid / B;

    const int l  = sel[b * 4 + s];
    const int pr = (l / nW) * 16;
    const int pc = (l % nW) * 16;

    for (int t = lane; t < 1024; t += 32)
        sD[(t >> 5) * LPITCH + (t & 31)] = Dg[t];
    const float* xp = x + (((size_t)b * C + c) * H + pr) * (size_t)Wd + pc;
    for (int t = lane; t < 1024; t += 32)
        sP[(t >> 5) * LPITCH + (t & 31)] = xp[(size_t)(t >> 5) * Wd + (t & 31)];
    __syncthreads();

    const int half = lane >> 4;
    const int n    = lane & 15;

    v8f a0[2][2], a1[2][2];
    gemm32(sD, false, sP, false, a0, lane);     // T = D * P
    store_cd(sT, a0, lane);
    __syncthreads();
    gemm32(sT, false, sD, true, a1, lane);      // Xdct = T * D^T

    // Apply LEVEL_FILT (identity for WS=32; computed faithfully).
    #pragma unroll
    for (int i = 0; i < 2; ++i)
        #pragma unroll
        for (int j = 0; j < 2; ++j)
            #pragma unroll
            for (int rr = 0; rr < 8; ++rr) {
                const int m  = i * 16 + half * 8 + rr;
                const int nn = j * 16 + n;
                const float f = (((m + nn) > 64) || ((m + nn) < 0)) ? 0.f : 1.f;
                a1[i][j][rr] *= f;
            }

    __syncthreads();                            // sP reads done (patch dead)
    store_cd(sP, a1, lane);                     // Xf -> sP
    __syncthreads();
    gemm32(sD, true, sP, false, a0, lane);      // U = D^T * Xf
    __syncthreads();                            // sT reads done
    store_cd(sT, a0, lane);
    __syncthreads();
    gemm32(sT, false, sD, false, a1, lane);     // Y = U * D   (= level_y)

    float* op = out + (((size_t)s * B + b) * C + c) * 1024;
    #pragma unroll
    for (int i = 0; i < 2; ++i)
        #pragma unroll
        for (int j = 0; j < 2; ++j)
            #pragma unroll
            for (int rr = 0; rr < 8; ++rr)
                op[(i * 16 + half * 8 + rr) * 32 + j * 16 + n] = a1[i][j][rr];
}

// ---------------------------------------------------------------------------
extern "C" void kernel_launch(void* const* d_in, const int* in_sizes, int n_in,
                              void* d_out, int out_size, void* d_ws, size_t ws_size,
                              hipStream_t stream)
{
    const float* x = (const float*)d_in[0];
    float* out = (float*)d_out;

    const int B = 16, C = 3, H = 512, Wd = 512;
    const int nH = (H - WSZ) / 16 + 1;          // 31
    const int nW = (Wd - WSZ) / 16 + 1;         // 31
    const int L  = nH * nW;                     // 961

    // Workspace layout (floats): D[1024] | W[1024] | grade[B*L] | sel[B*4] ints
    float* ws    = (float*)d_ws;
    float* Dg    = ws;
    float* Wg    = ws + 1024;
    float* grade = ws + 2048;
    int*   sel   = (int*)(ws + 2048 + B * L);

    hipLaunchKernelGGL(init_consts_kernel, dim3(1), dim3(1024), 0, stream, Dg, Wg);
    hipLaunchKernelGGL(dct_grade_kernel, dim3(B * L), dim3(32), 0, stream,
                       x, Dg, Wg, grade, C, H, Wd, nW, L);
    hipLaunchKernelGGL(select_kernel, dim3(B), dim3(256), 0, stream, grade, sel, L);
    hipLaunchKernelGGL(level_out_kernel, dim3(B * 4 * C), dim3(32), 0, stream,
                       x, Dg, sel, out, B, C, H, Wd, nW);
}